// KabschLoss_79370995630393
// MI455X (gfx1250) — compile-verified
//
#include <hip/hip_runtime.h>
#include <math.h>

#define BATCHES 8192
#define NPTS    1024

typedef __attribute__((ext_vector_type(2))) float v2f;
typedef __attribute__((ext_vector_type(8))) float v8f;

// --------------------------------------------------------------------------
// Kernel A: per-batch 8x8 Gram matrices via V_WMMA_F32_16X16X4_F32.
// Augmented point feature vector z = (x0,x1,x2,1, y0,y1,y2,1).
// G_b = sum_n z z^T. Two batches per wave packed into the 16x16 tile:
//   rows/cols 0-7  -> batch b0, rows/cols 8-15 -> batch b1.
// Gram => A fragment == B fragment. K=4 points per WMMA, 256 steps.
// --------------------------------------------------------------------------
__global__ __launch_bounds__(256) void kabsch_gram_wmma(
    const float* __restrict__ x, const float* __restrict__ y,
    float* __restrict__ gws)
{
  const int lane = threadIdx.x & 31;
  const int wave = blockIdx.x * (blockDim.x >> 5) + (threadIdx.x >> 5);
  const int b0 = wave * 2;
  if (b0 >= BATCHES) return;   // wave-uniform guard

  // A-fragment mapping for 16x4 f32 (ISA 7.12.2):
  //   lanes 0-15:  VGPR0=A[row][0], VGPR1=A[row][1]
  //   lanes 16-31: VGPR0=A[row][2], VGPR1=A[row][3]
  const int row = lane & 15;          // tile row (== tile col for B fragment)
  const int kb  = (lane >> 4) * 2;    // k-slot base {0,2}
  const int bb  = b0 + (row >> 3);    // which batch this row belongs to
  const int f   = row & 7;            // feature index 0..7
  const bool is_one = (f == 3) || (f == 7);

  // Constant-1 lanes alias the coord-0 address of the same window (no extra
  // traffic, keeps loads unconditional so EXEC stays all-ones for WMMA).
  const float* srcp;
  if (f < 4) srcp = x + (size_t)bb * (NPTS * 3) + (f < 3 ? f : 0);
  else       srcp = y + (size_t)bb * (NPTS * 3) + (f < 7 ? (f - 4) : 0);
  srcp += kb * 3;

  v8f acc = {};
  #pragma unroll 8
  for (int s = 0; s < NPTS / 4; ++s) {
    float a0 = srcp[s * 12];          // point 4s+kb   , this feature
    float a1 = srcp[s * 12 + 3];      // point 4s+kb+1 , this feature
    v2f ab;
    ab.x = is_one ? 1.0f : a0;
    ab.y = is_one ? 1.0f : a1;
    // D = A*B + C ; A == B (Gram)
    acc = __builtin_amdgcn_wmma_f32_16x16x4_f32(
        false, ab, false, ab, (short)0, acc, false, false);
  }

  // C/D layout: VGPR r, lanes 0-15 -> (M=r, N=lane); lanes 16-31 -> (M=r+8, N=lane-16)
  // batch0 block (rows 0-7, cols 0-7): lanes 0-7 hold column `lane`, rows acc[0..7]
  // batch1 block (rows 8-15, cols 8-15): lanes 24-31 hold column `lane-24`
  float* dst = nullptr;
  if (lane < 8)        dst = gws + (size_t)b0 * 64 + lane * 8;
  else if (lane >= 24) dst = gws + (size_t)(b0 + 1) * 64 + (lane - 24) * 8;
  if (dst) {
    float4 lo = make_float4(acc[0], acc[1], acc[2], acc[3]);
    float4 hi = make_float4(acc[4], acc[5], acc[6], acc[7]);
    ((float4*)dst)[0] = lo;          // gws[batch][col][row], row-contiguous
    ((float4*)dst)[1] = hi;
  }
}

// --------------------------------------------------------------------------
// Kernel B: per-batch centering + 3x3 C^T C eigenvalues (trigonometric
// closed form) -> sum of singular values -> per-batch loss contribution.
// Block-reduced partial sums (deterministic, no float atomics).
// --------------------------------------------------------------------------
__global__ __launch_bounds__(256) void kabsch_svd_loss(
    const float* __restrict__ gws, float* __restrict__ partial)
{
  const int b = blockIdx.x * blockDim.x + threadIdx.x;   // grid sized exactly
  float Gr[64];
  const float4* gv = (const float4*)(gws + (size_t)b * 64);
  #pragma unroll
  for (int i = 0; i < 16; ++i) ((float4*)Gr)[i] = gv[i];
  #define G_(r, c) Gr[(c) * 8 + (r)]

  const float invN = 1.0f / (float)NPTS;
  float Sx[3] = {G_(0, 3), G_(1, 3), G_(2, 3)};
  float Sy[3] = {G_(4, 3), G_(5, 3), G_(6, 3)};

  float Cm[3][3];
  #pragma unroll
  for (int i = 0; i < 3; ++i)
    #pragma unroll
    for (int j = 0; j < 3; ++j)
      Cm[i][j] = G_(i, 4 + j) - Sx[i] * Sy[j] * invN;

  float ex = G_(0, 0) + G_(1, 1) + G_(2, 2)
           - (Sx[0]*Sx[0] + Sx[1]*Sx[1] + Sx[2]*Sx[2]) * invN;
  float ey = G_(4, 4) + G_(5, 5) + G_(6, 6)
           - (Sy[0]*Sy[0] + Sy[1]*Sy[1] + Sy[2]*Sy[2]) * invN;

  // A = Cm^T Cm (symmetric PSD); eigenvalues -> squared singular values
  float a00 = Cm[0][0]*Cm[0][0] + Cm[1][0]*Cm[1][0] + Cm[2][0]*Cm[2][0];
  float a11 = Cm[0][1]*Cm[0][1] + Cm[1][1]*Cm[1][1] + Cm[2][1]*Cm[2][1];
  float a22 = Cm[0][2]*Cm[0][2] + Cm[1][2]*Cm[1][2] + Cm[2][2]*Cm[2][2];
  float a01 = Cm[0][0]*Cm[0][1] + Cm[1][0]*Cm[1][1] + Cm[2][0]*Cm[2][1];
  float a02 = Cm[0][0]*Cm[0][2] + Cm[1][0]*Cm[1][2] + Cm[2][0]*Cm[2][2];
  float a12 = Cm[0][1]*Cm[0][2] + Cm[1][1]*Cm[1][2] + Cm[2][1]*Cm[2][2];

  float p1 = a01*a01 + a02*a02 + a12*a12;
  float q  = (a00 + a11 + a22) * (1.0f / 3.0f);
  float d0 = a00 - q, d1 = a11 - q, d2 = a22 - q;
  float p2 = d0*d0 + d1*d1 + d2*d2 + 2.0f * p1;

  float ssum;
  if (p2 <= 1e-24f) {
    ssum = 3.0f * sqrtf(fmaxf(q, 0.0f));
  } else {
    float p  = sqrtf(p2 * (1.0f / 6.0f));
    float ip = 1.0f / p;
    float b00 = d0 * ip, b11 = d1 * ip, b22 = d2 * ip;
    float b01 = a01 * ip, b02 = a02 * ip, b12 = a12 * ip;
    float detB = b00 * (b11 * b22 - b12 * b12)
               - b01 * (b01 * b22 - b12 * b02)
               + b02 * (b01 * b12 - b11 * b02);
    float r = fminf(1.0f, fmaxf(-1.0f, 0.5f * detB));
    float phi = acosf(r) * (1.0f / 3.0f);
    float e0 = q + 2.0f * p * cosf(phi);
    float e2 = q + 2.0f * p * cosf(phi + 2.0943951023931953f); // + 2*pi/3
    float e1 = 3.0f * q - e0 - e2;
    ssum = sqrtf(fmaxf(e0, 0.0f)) + sqrtf(fmaxf(e1, 0.0f)) + sqrtf(fmaxf(e2, 0.0f));
  }

  float contrib = ex + ey - 2.0f * ssum;

  __shared__ float red[256];
  red[threadIdx.x] = contrib;
  __syncthreads();
  #pragma unroll
  for (int s = 128; s > 0; s >>= 1) {
    if ((int)threadIdx.x < s) red[threadIdx.x] += red[threadIdx.x + s];
    __syncthreads();
  }
  if (threadIdx.x == 0) partial[blockIdx.x] = red[0];
}

// --------------------------------------------------------------------------
// Kernel C: final deterministic reduction of block partials -> scalar mean.
// --------------------------------------------------------------------------
__global__ void kabsch_final(const float* __restrict__ partial,
                             float* __restrict__ out, int n)
{
  float v = 0.0f;
  for (int i = threadIdx.x; i < n; i += 32) v += partial[i];
  #pragma unroll
  for (int off = 16; off > 0; off >>= 1) v += __shfl_down(v, off, 32);
  if (threadIdx.x == 0)
    out[0] = v * (1.0f / ((float)BATCHES * (float)NPTS * 3.0f));
}

extern "C" void kernel_launch(void* const* d_in, const int* in_sizes, int n_in,
                              void* d_out, int out_size, void* d_ws, size_t ws_size,
                              hipStream_t stream) {
  const float* x = (const float*)d_in[0];
  const float* y = (const float*)d_in[1];
  float* out = (float*)d_out;
  float* gws = (float*)d_ws;                       // 8192 * 64 floats = 2 MiB
  float* partial = gws + (size_t)BATCHES * 64;     // + 32 floats

  // 4096 waves (2 batches each), 8 waves per block -> 512 blocks
  kabsch_gram_wmma<<<BATCHES / 2 / 8, 256, 0, stream>>>(x, y, gws);
  kabsch_svd_loss<<<BATCHES / 256, 256, 0, stream>>>(gws, partial);
  kabsch_final<<<1, 32, 0, stream>>>(partial, out, BATCHES / 256);
}